// GCNLayer_56341380989305
// MI455X (gfx1250) — compile-verified
//
#include <hip/hip_runtime.h>

typedef __attribute__((ext_vector_type(2))) float v2f;
typedef __attribute__((ext_vector_type(8))) float v8f;

#define D_FEAT 128

// ---------------------------------------------------------------------------
// Kernel 1: zero the h_neigh accumulator in d_ws (float4 stores, coalesced)
// ---------------------------------------------------------------------------
__global__ void gcn_zero_kernel(float4* __restrict__ p, long n4) {
    long i = (long)blockIdx.x * blockDim.x + threadIdx.x;
    if (i < n4) p[i] = make_float4(0.f, 0.f, 0.f, 0.f);
}

// ---------------------------------------------------------------------------
// Kernel 2: edge scatter-add. One wave32 per edge: lane l covers columns
// [4l, 4l+4) of the 128-wide feature row. float4 gather from feature[src],
// 4 non-returning global fp32 atomics into acc[dst]. Both arrays are
// L2-resident on MI455X (51 MB each vs 192 MB L2).
// ---------------------------------------------------------------------------
__global__ void gcn_scatter_kernel(const float* __restrict__ feat,
                                   const int* __restrict__ src,
                                   const int* __restrict__ dst,
                                   float* __restrict__ acc,
                                   int n_edges) {
    long t = (long)blockIdx.x * blockDim.x + threadIdx.x;
    int e = (int)(t >> 5);
    if (e >= n_edges) return;
    int c = ((int)t & 31) << 2;           // column chunk: lane*4

    int s = src[e];
    int d = dst[e];

    const float4 v = *(const float4*)(feat + (size_t)s * D_FEAT + c);
    float* out = acc + (size_t)d * D_FEAT + c;
    atomicAdd(out + 0, v.x);              // global_atomic_add_f32, no return
    atomicAdd(out + 1, v.y);
    atomicAdd(out + 2, v.z);
    atomicAdd(out + 3, v.w);
}

// ---------------------------------------------------------------------------
// Kernel 3: h = acc @ W^T + b  via V_WMMA_F32_16X16X4_F32 (fp32 exact).
// Each wave computes a 16x64 output strip (4 column tiles): the A fragment
// is loaded once per K-step and reused by 4 WMMAs (1.25 b64 loads / WMMA),
// giving 4 independent accumulator chains for ILP across the matrix pipe.
//
// Fragment layouts (ISA 7.12.2):
//   A 16x4 f32 : lane L holds A[row0 + (L&15)][k + koff .. +1], koff = 2*(L>=16)
//   B 4x16 f32 : lane L holds Wt[k + koff .. +1][col + (L&15)]
//                = W[col+(L&15)][k+koff .. +1]   (contiguous float2 in W)
//   C/D 16x16  : VGPR r -> M = row0 + r + 8*(L>=16), N = col + (L&15)
// ---------------------------------------------------------------------------
__global__ void gcn_wmma_gemm_kernel(const float* __restrict__ A,   // [N,128]
                                     const float* __restrict__ W,   // [128,128] row-major [out][in]
                                     const float* __restrict__ bias,// [128]
                                     float* __restrict__ out,       // [N,128]
                                     int n_nodes) {
    const int lane = threadIdx.x & 31;
    const int wave = threadIdx.x >> 5;
    const int strip = blockIdx.x * (blockDim.x >> 5) + wave;

    const int rowTile = strip >> 1;       // 2 strips of 64 cols per row tile
    const int row0 = rowTile << 4;
    if (row0 >= n_nodes) return;
    const int col0 = (strip & 1) << 6;    // 0 or 64

    const int half = lane >> 4;           // 0: lanes 0-15, 1: lanes 16-31
    const int l15  = lane & 15;
    const int koff = half << 1;           // K sub-offset within the step of 4

    // clamp A row for ragged tail (stores are guarded below)
    int arow = row0 + l15;
    if (arow >= n_nodes) arow = n_nodes - 1;

    const float* Aptr = A + (size_t)arow * D_FEAT + koff;
    const float* Wptr = W + (size_t)(col0 + l15) * D_FEAT + koff;

    v8f c0 = {}, c1 = {}, c2 = {}, c3 = {};

#pragma unroll 4
    for (int k = 0; k < D_FEAT; k += 4) {
        const v2f a = *(const v2f*)(Aptr + k);
        const v2f b0 = *(const v2f*)(Wptr + k);                 // col tile 0
        const v2f b1 = *(const v2f*)(Wptr + k + 16 * D_FEAT);   // col tile 1
        const v2f b2 = *(const v2f*)(Wptr + k + 32 * D_FEAT);   // col tile 2
        const v2f b3 = *(const v2f*)(Wptr + k + 48 * D_FEAT);   // col tile 3
        c0 = __builtin_amdgcn_wmma_f32_16x16x4_f32(
                 false, a, false, b0, (short)0, c0, false, false);
        c1 = __builtin_amdgcn_wmma_f32_16x16x4_f32(
                 false, a, false, b1, (short)0, c1, false, false);
        c2 = __builtin_amdgcn_wmma_f32_16x16x4_f32(
                 false, a, false, b2, (short)0, c2, false, false);
        c3 = __builtin_amdgcn_wmma_f32_16x16x4_f32(
                 false, a, false, b3, (short)0, c3, false, false);
    }

    const float bv0 = bias[col0 + l15];
    const float bv1 = bias[col0 + 16 + l15];
    const float bv2 = bias[col0 + 32 + l15];
    const float bv3 = bias[col0 + 48 + l15];

#pragma unroll
    for (int r = 0; r < 8; ++r) {
        int row = row0 + r + (half << 3);
        if (row < n_nodes) {
            float* o = out + (size_t)row * D_FEAT + col0 + l15;
            o[0]  = c0[r] + bv0;
            o[16] = c1[r] + bv1;
            o[32] = c2[r] + bv2;
            o[48] = c3[r] + bv3;
        }
    }
}

// ---------------------------------------------------------------------------
// Launch: zero ws -> scatter-add -> WMMA GEMM (all on `stream`, no sync)
// ---------------------------------------------------------------------------
extern "C" void kernel_launch(void* const* d_in, const int* in_sizes, int n_in,
                              void* d_out, int out_size, void* d_ws, size_t ws_size,
                              hipStream_t stream) {
    const float* feat = (const float*)d_in[0];   // [N,128]
    const int*   src  = (const int*)d_in[1];     // [E]
    const int*   dst  = (const int*)d_in[2];     // [E]
    const float* W    = (const float*)d_in[3];   // [128,128]
    const float* bias = (const float*)d_in[4];   // [128]

    const int n_edges = in_sizes[1];
    const int n_nodes = in_sizes[0] / D_FEAT;

    float* acc = (float*)d_ws;                   // [N,128] fp32 accumulator
    float* out = (float*)d_out;

    // 1) zero accumulator (needed every call for determinism)
    {
        long n4 = (long)n_nodes * (D_FEAT / 4);
        long blocks = (n4 + 255) / 256;
        gcn_zero_kernel<<<(int)blocks, 256, 0, stream>>>((float4*)acc, n4);
    }

    // 2) edge scatter-add: one wave32 per edge
    {
        long threads = (long)n_edges * 32;
        long blocks = (threads + 255) / 256;
        gcn_scatter_kernel<<<(int)blocks, 256, 0, stream>>>(feat, src, dst,
                                                            acc, n_edges);
    }

    // 3) WMMA fp32 GEMM + bias (each wave: 16 rows x 64 cols)
    {
        int rowTiles = (n_nodes + 15) / 16;
        int strips = rowTiles * 2;               // 2 x 64-col strips per row tile
        int wavesPerBlock = 8;                   // 256 threads
        int blocks = (strips + wavesPerBlock - 1) / wavesPerBlock;
        gcn_wmma_gemm_kernel<<<blocks, 256, 0, stream>>>(acc, W, bias,
                                                         out, n_nodes);
    }
}